// CarafeDownsample_70746701299867
// MI455X (gfx1250) — compile-verified
//
#include <hip/hip_runtime.h>

typedef float v2f __attribute__((ext_vector_type(2)));
typedef float v8f __attribute__((ext_vector_type(8)));

#define B_N   8
#define C_IN  256
#define M_CH  64
#define H_IMG 128
#define W_IMG 128
#define HW    (H_IMG * W_IMG)   // 16384
#define KK2   25                // 5*5
#define nH    64
#define nW    64
#define nHW   (nH * nW)         // 4096
#define K2DIM (M_CH * 9)        // 576

// ---------------------------------------------------------------------------
// Kernel 1: 1x1 conv (C=256 -> M=64) as GEMM with V_WMMA_F32_16X16X4_F32.
// Each wave computes a 64(M) x 16(pixel) stripe -> x streamed once from HBM.
// A (w1) staged in LDS (64 KB), read as ds_load_b64 per K-step.
// ---------------------------------------------------------------------------
__global__ __launch_bounds__(128) void k1_compress(
    const float* __restrict__ x, const float* __restrict__ w1,
    const float* __restrict__ b1, float* __restrict__ ker1)
{
    __shared__ float lw1[M_CH * C_IN];   // 64 KB
    const int tid = threadIdx.x;
    {
        const float4* src = (const float4*)w1;
        float4* dst = (float4*)lw1;
        #pragma unroll
        for (int i = 0; i < (M_CH * C_IN / 4) / 128; ++i)   // 32 iters
            dst[tid + i * 128] = src[tid + i * 128];
    }
    __syncthreads();

    const int lane = tid & 31;
    const int wv   = tid >> 5;          // wave in block: 0..3
    const int half = lane >> 4;         // 0/1 (K split per ISA A/B layout)
    const int l15  = lane & 15;

    const int b   = blockIdx.x >> 8;    // 256 n-blocks per batch
    const int nb  = blockIdx.x & 255;
    const int col = nb * 64 + wv * 16 + l15;       // pixel index in [0,16384)
    const float* xb = x + (size_t)b * C_IN * HW;

    v8f a0 = {0,0,0,0,0,0,0,0};
    v8f a1 = a0, a2 = a0, a3 = a0;

    for (int k = 0; k < C_IN; k += 4) {
        const int kk = k + 2 * half;
        // B 4x16: lanes 0-15 -> K=k..k+1 (vgpr0/1), lanes 16-31 -> K=k+2..k+3
        v2f bm;
        bm.x = xb[(size_t)kk * HW + col];
        bm.y = xb[(size_t)kk * HW + HW + col];
        // A 16x4: M = base + l15, K = kk, kk+1 (float2 from LDS, 8B aligned)
        v2f am0 = *(const v2f*)&lw1[( 0 + l15) * C_IN + kk];
        v2f am1 = *(const v2f*)&lw1[(16 + l15) * C_IN + kk];
        v2f am2 = *(const v2f*)&lw1[(32 + l15) * C_IN + kk];
        v2f am3 = *(const v2f*)&lw1[(48 + l15) * C_IN + kk];
        a0 = __builtin_amdgcn_wmma_f32_16x16x4_f32(false, am0, false, bm, (short)0, a0, false, false);
        a1 = __builtin_amdgcn_wmma_f32_16x16x4_f32(false, am1, false, bm, (short)0, a1, false, false);
        a2 = __builtin_amdgcn_wmma_f32_16x16x4_f32(false, am2, false, bm, (short)0, a2, false, false);
        a3 = __builtin_amdgcn_wmma_f32_16x16x4_f32(false, am3, false, bm, (short)0, a3, false, false);
    }

    // C/D layout: vgpr v, lanes 0-15 -> M=v, lanes 16-31 -> M=v+8; N = l15
    float* ob = ker1 + (size_t)b * M_CH * HW;
    #pragma unroll
    for (int v = 0; v < 8; ++v) {
        const int m = v + 8 * half;
        ob[(size_t)(m     ) * HW + col] = a0[v] + b1[m];
        ob[(size_t)(m + 16) * HW + col] = a1[v] + b1[m + 16];
        ob[(size_t)(m + 32) * HW + col] = a2[v] + b1[m + 32];
        ob[(size_t)(m + 48) * HW + col] = a3[v] + b1[m + 48];
    }
}

// ---------------------------------------------------------------------------
// Kernel 2: 3x3 stride-2 conv (64 -> 25) as im2col GEMM (M padded 25->32)
// with WMMA f32, fused softmax over the 25 channels per output pixel.
// ---------------------------------------------------------------------------
__global__ __launch_bounds__(128) void k2_encoder(
    const float* __restrict__ ker1, const float* __restrict__ w2,
    const float* __restrict__ b2, float* __restrict__ wgt)
{
    __shared__ float tile[4][32][16];   // per-wave 32xM x 16xN result (8 KB)

    const int tid  = threadIdx.x;
    const int lane = tid & 31;
    const int wv   = tid >> 5;
    const int half = lane >> 4;
    const int l15  = lane & 15;

    const int p  = (blockIdx.x * 4 + wv) * 16 + l15;   // pixel id 0..32767
    const int b  = p >> 12;
    const int oh = (p >> 6) & 63;
    const int ow = p & 63;
    const float* kb = ker1 + (size_t)b * M_CH * HW;

    v8f a0 = {0,0,0,0,0,0,0,0};
    v8f a1 = a0;

    for (int k = 0; k < K2DIM; k += 4) {
        const int kk = k + 2 * half;
        // B: im2col element pair (K = kk, kk+1) for column = this lane's pixel
        v2f bm;
        {
            const int c = kk / 9, r = kk - c * 9;
            const int dy = r / 3, dx = r - dy * 3;
            const int ih = 2 * oh - 1 + dy, iw = 2 * ow - 1 + dx;
            const bool ok = ((unsigned)ih < 128u) && ((unsigned)iw < 128u);
            bm.x = ok ? kb[(size_t)c * HW + ih * W_IMG + iw] : 0.f;
        }
        {
            const int kq = kk + 1;
            const int c = kq / 9, r = kq - c * 9;
            const int dy = r / 3, dx = r - dy * 3;
            const int ih = 2 * oh - 1 + dy, iw = 2 * ow - 1 + dx;
            const bool ok = ((unsigned)ih < 128u) && ((unsigned)iw < 128u);
            bm.y = ok ? kb[(size_t)c * HW + ih * W_IMG + iw] : 0.f;
        }
        // A: rows of w2 (25x576), rows 25..31 are zero padding
        v2f am0 = *(const v2f*)&w2[(size_t)l15 * K2DIM + kk];
        v2f am1;
        const int m1 = 16 + l15;
        if (m1 < KK2) am1 = *(const v2f*)&w2[(size_t)m1 * K2DIM + kk];
        else { am1.x = 0.f; am1.y = 0.f; }

        a0 = __builtin_amdgcn_wmma_f32_16x16x4_f32(false, am0, false, bm, (short)0, a0, false, false);
        a1 = __builtin_amdgcn_wmma_f32_16x16x4_f32(false, am1, false, bm, (short)0, a1, false, false);
    }

    // Transpose result through LDS so one lane owns all 25 logits of a pixel
    #pragma unroll
    for (int v = 0; v < 8; ++v) {
        tile[wv][v + 8 * half][l15]      = a0[v];
        tile[wv][16 + v + 8 * half][l15] = a1[v];
    }
    __syncthreads();

    if (lane < 16) {
        float vals[KK2];
        float mx = -1e30f;
        #pragma unroll
        for (int kc = 0; kc < KK2; ++kc) {
            vals[kc] = tile[wv][kc][l15] + b2[kc];
            mx = fmaxf(mx, vals[kc]);
        }
        float s = 0.f;
        #pragma unroll
        for (int kc = 0; kc < KK2; ++kc) { vals[kc] = __expf(vals[kc] - mx); s += vals[kc]; }
        const float inv = 1.f / s;
        float* wb = wgt + (size_t)b * KK2 * nHW + oh * nW + ow;
        #pragma unroll
        for (int kc = 0; kc < KK2; ++kc)
            wb[(size_t)kc * nHW] = vals[kc] * inv;
    }
}

// ---------------------------------------------------------------------------
// Kernel 3: CARAFE reassembly. Block = (b, oh, 16 ow's). Input footprint
// (5 rows x 35 cols) staged in LDS per 16-channel chunk -> x read once.
// Softmax weights held in registers; LDS reads as aligned b64 pairs.
// ---------------------------------------------------------------------------
__global__ __launch_bounds__(256) void k3_reassemble(
    const float* __restrict__ x, const float* __restrict__ wgt,
    float* __restrict__ out)
{
    __shared__ float xt[16][5][36];     // 11.25 KB (padded rows, 8B aligned)
    __shared__ float wt[KK2][16];       // 1.6 KB

    const int tid = threadIdx.x;
    const int b   = blockIdx.x >> 8;
    const int oh  = (blockIdx.x >> 2) & 63;
    const int ow0 = (blockIdx.x & 3) * 16;

    for (int i = tid; i < KK2 * 16; i += 256) {
        const int kc = i >> 4, px = i & 15;
        wt[kc][px] = wgt[((size_t)b * KK2 + kc) * nHW + oh * nW + ow0 + px];
    }
    __syncthreads();

    const int px = tid & 15;
    const int ch = tid >> 4;

    float wr[KK2];
    #pragma unroll
    for (int kc = 0; kc < KK2; ++kc) wr[kc] = wt[kc][px];

    const float* xb = x + (size_t)b * C_IN * HW;
    float* ob = out + (size_t)b * C_IN * nHW + oh * nW + ow0 + px;

    for (int cc = 0; cc < C_IN / 16; ++cc) {
        for (int idx = tid; idx < 16 * 5 * 35; idx += 256) {
            const int lch = idx / 175;
            const int rem = idx - lch * 175;
            const int r = rem / 35;
            const int j = rem - r * 35;
            const int ih = 2 * oh - 2 + r;
            const int ic = 2 * ow0 - 2 + j;
            const bool ok = ((unsigned)ih < 128u) && ((unsigned)ic < 128u);
            xt[lch][r][j] = ok ? xb[(size_t)(cc * 16 + lch) * HW + ih * W_IMG + ic] : 0.f;
        }
        __syncthreads();

        float acc = 0.f;
        #pragma unroll
        for (int r = 0; r < 5; ++r) {
            const v2f* rp = (const v2f*)&xt[ch][r][2 * px];   // 8B aligned
            const v2f p01 = rp[0];
            const v2f p23 = rp[1];
            const float p4 = xt[ch][r][2 * px + 4];
            acc += p01.x * wr[r * 5 + 0] + p01.y * wr[r * 5 + 1]
                 + p23.x * wr[r * 5 + 2] + p23.y * wr[r * 5 + 3]
                 + p4    * wr[r * 5 + 4];
        }
        ob[(size_t)(cc * 16 + ch) * nHW] = acc;
        __syncthreads();
    }
}

// ---------------------------------------------------------------------------
extern "C" void kernel_launch(void* const* d_in, const int* in_sizes, int n_in,
                              void* d_out, int out_size, void* d_ws, size_t ws_size,
                              hipStream_t stream) {
    (void)in_sizes; (void)n_in; (void)out_size; (void)ws_size;
    const float* x  = (const float*)d_in[0];
    const float* w1 = (const float*)d_in[1];
    const float* b1 = (const float*)d_in[2];
    const float* w2 = (const float*)d_in[3];
    const float* b2 = (const float*)d_in[4];
    float* out  = (float*)d_out;

    float* ker1 = (float*)d_ws;                       // 8*64*128*128 floats (33.5 MB)
    float* wgt  = ker1 + (size_t)B_N * M_CH * HW;     // 8*25*64*64 floats  (3.3 MB)

    k1_compress  <<<dim3(B_N * 256), dim3(128), 0, stream>>>(x, w1, b1, ker1);
    k2_encoder   <<<dim3(512),       dim3(128), 0, stream>>>(ker1, w2, b2, wgt);
    k3_reassemble<<<dim3(2048),      dim3(256), 0, stream>>>(x, wgt, out);
}